// GIN_10213432229988
// MI455X (gfx1250) — compile-verified
//
#include <hip/hip_runtime.h>
#include <hip/hip_bf16.h>

typedef __attribute__((ext_vector_type(2))) float v2f;
typedef __attribute__((ext_vector_type(8))) float v8f;

#define NN 50000
#define NE 800000
#define NG 500
#define NF 128
#define NH 64
#define BN_EPS 1e-5f

// ---------------- utility kernels ----------------

__global__ void zero_k(float* __restrict__ p, long n) {
    long i = (long)blockIdx.x * blockDim.x + threadIdx.x;
    if (i < n) p[i] = 0.f;
}

__global__ void count_k(const int* __restrict__ batch, float* __restrict__ cnt, int n) {
    int i = blockIdx.x * blockDim.x + threadIdx.x;
    if (i < n) atomicAdd(&cnt[batch[i]], 1.0f);
}

// ---------------- WMMA fp32 GEMM: Y[M,64] = (X (+X2)) @ W[K,64] (+bias, relu) ----------------
// One wave computes a 16x16 tile with K/4 fully-unrolled V_WMMA_F32_16X16X4_F32 steps.
// Block = 256 threads = 8 waves: 2 row tiles x 4 col tiles = 32x64 per block.
template <int K, bool HAS_X2>
__global__ void gemm_wmma(const float* __restrict__ X, const float* __restrict__ X2,
                          const float* __restrict__ W, const float* __restrict__ bias,
                          float* __restrict__ Y, int M, int do_relu) {
    int wave = threadIdx.x >> 5;
    int lane = threadIdx.x & 31;
    int rowTile = blockIdx.x * 2 + (wave >> 2);
    int m0 = rowTile << 4;
    if (m0 >= M) return;                    // uniform per wave -> EXEC stays all-1s
    int n0 = (wave & 3) << 4;
    int half = lane >> 4;                   // selects K pair {0,1} vs {2,3}
    int l16 = lane & 15;

    const float* xr  = X + (size_t)(m0 + l16) * K;
    const float* x2r = HAS_X2 ? (X2 + (size_t)(m0 + l16) * K) : nullptr;
    const float* wc  = W + n0 + l16;        // column base of weight matrix

    v8f c = {0.f, 0.f, 0.f, 0.f, 0.f, 0.f, 0.f, 0.f};

#pragma unroll
    for (int kk = 0; kk < K; kk += 4) {
        const int kc = kk + (half << 1);
        // A fragment: row m0+l16, cols kc..kc+1 (ISA 16x4 f32 layout) -> one b64 load
        v2f a = *(const v2f*)(xr + kc);
        if (HAS_X2) {
            v2f a2 = *(const v2f*)(x2r + kc);
            a.x += a2.x; a.y += a2.y;
        }
        // B fragment: W rows kc, kc+1 at col n0+l16 (4x16 f32 layout)
        v2f b;
        b.x = wc[(size_t)kc * NH];
        b.y = wc[(size_t)(kc + 1) * NH];
        c = __builtin_amdgcn_wmma_f32_16x16x4_f32(false, a, false, b,
                                                  (short)0, c, false, false);
    }

    int ocol = n0 + l16;
    float bv = bias ? bias[ocol] : 0.f;
    int orow = m0 + (half << 3);
#pragma unroll
    for (int v = 0; v < 8; ++v) {
        float val = c[v] + bv;
        if (do_relu) val = fmaxf(val, 0.f);
        Y[(size_t)(orow + v) * NH + ocol] = val;
    }
}

// ---------------- per-channel sum / sumsq over rows (for BatchNorm) ----------------
// stats[0..63] = sum, stats[64..127] = sumsq  (must be pre-zeroed)
__global__ void colstats_k(const float* __restrict__ Y, int M, float* __restrict__ stats) {
    __shared__ float ls[256];
    __shared__ float lq[256];
    int c  = threadIdx.x & 63;
    int rg = threadIdx.x >> 6;   // 4 row-groups per block
    float s = 0.f, q = 0.f;
    for (int i = blockIdx.x * 4 + rg; i < M; i += gridDim.x * 4) {
        float v = Y[(size_t)i * NH + c];
        s += v; q += v * v;
    }
    ls[threadIdx.x] = s; lq[threadIdx.x] = q;
    __syncthreads();
    if (rg == 0) {
        s = ls[c] + ls[c + 64] + ls[c + 128] + ls[c + 192];
        q = lq[c] + lq[c + 64] + lq[c + 128] + lq[c + 192];
        atomicAdd(&stats[c], s);
        atomicAdd(&stats[64 + c], q);
    }
}

// scsh[0..63]=scale, scsh[64..127]=shift    (launch <<<1,64>>>)
__global__ void bn_prep_k(const float* __restrict__ stats, const float* __restrict__ gamma,
                          const float* __restrict__ beta, float* __restrict__ scsh, float invM) {
    int c = threadIdx.x;
    float mean = stats[c] * invM;
    float var  = stats[64 + c] * invM - mean * mean;
    float sc   = gamma[c] * rsqrtf(var + BN_EPS);
    scsh[c]      = sc;
    scsh[64 + c] = beta[c] - mean * sc;
}

// in-place BN apply + graph pool-sum accumulation
__global__ void bn_apply_pool_k(float* __restrict__ Y, const int* __restrict__ batch,
                                const float* __restrict__ scsh, float* __restrict__ psum, long n) {
    long gid = (long)blockIdx.x * blockDim.x + threadIdx.x;
    if (gid >= n) return;
    int c = (int)(gid & 63);
    int node = (int)(gid >> 6);
    float v = Y[gid] * scsh[c] + scsh[64 + c];
    Y[gid] = v;
    atomicAdd(&psum[(size_t)batch[node] * NH + c], v);
}

// psum -> mean (in place) and write embeds slice to output
__global__ void pool_mean_fin_k(float* __restrict__ psum, const float* __restrict__ cnt,
                                float* __restrict__ out_embed) {
    int j = blockIdx.x * blockDim.x + threadIdx.x;
    if (j >= NG * NH) return;
    float m = psum[j] / fmaxf(cnt[j >> 6], 1.f);
    psum[j] = m;
    out_embed[j] = m;
}

__global__ void pool_sq_k(const float* __restrict__ Y, const int* __restrict__ batch,
                          const float* __restrict__ pmean, float* __restrict__ psq, long n) {
    long gid = (long)blockIdx.x * blockDim.x + threadIdx.x;
    if (gid >= n) return;
    int c = (int)(gid & 63);
    int node = (int)(gid >> 6);
    int g = batch[node];
    float d = Y[gid] - pmean[(size_t)g * NH + c];
    atomicAdd(&psq[(size_t)g * NH + c], d * d);
}

__global__ void pool_std_fin_k(const float* __restrict__ psq, const float* __restrict__ cnt,
                               float* __restrict__ out_std) {
    int j = blockIdx.x * blockDim.x + threadIdx.x;
    if (j >= NG * NH) return;
    out_std[j] = sqrtf(psq[j] / fmaxf(cnt[j >> 6], 1.f));
}

// ---------------- GIN edge scatter: agg[dst] += h[src], 64 channels/edge ----------------
__global__ void scatter_add_k(const float* __restrict__ H, const int* __restrict__ src,
                              const int* __restrict__ dst, float* __restrict__ agg, long n) {
    long gid = (long)blockIdx.x * blockDim.x + threadIdx.x;
    if (gid >= n) return;
    int c = (int)(gid & 63);
    long e = gid >> 6;
    atomicAdd(&agg[(size_t)dst[e] * NH + c], H[(size_t)src[e] * NH + c]);
}

// ---------------- driver ----------------

extern "C" void kernel_launch(void* const* d_in, const int* in_sizes, int n_in,
                              void* d_out, int out_size, void* d_ws, size_t ws_size,
                              hipStream_t stream) {
    const float* x      = (const float*)d_in[0];
    const int*   ei     = (const int*)d_in[1];
    const int*   batch  = (const int*)d_in[2];
    const float* Wt     = (const float*)d_in[3];
    const float* bt     = (const float*)d_in[4];
    const float* gamma0 = (const float*)d_in[5];
    const float* beta0  = (const float*)d_in[6];
    const float* W1     = (const float*)d_in[7];
    const float* W2     = (const float*)d_in[8];
    const float* gammas = (const float*)d_in[9];
    const float* betas  = (const float*)d_in[10];
    float* out = (float*)d_out;

    const int*   src = ei;
    const int*   dst = ei + NE;

    const size_t NNH = (size_t)NN * NH;          // 3.2M floats per node-feature buffer
    float* ws = (float*)d_ws;
    float* bufA  = ws;                            // current features / gemm2 output (in-place BN)
    float* bufB  = bufA + NNH;                    // agg
    float* bufC  = bufB + NNH;                    // relu(z @ W1)
    float* psum  = bufC + NNH;                    // [NG*NH]
    float* psq   = psum + (size_t)NG * NH;        // [NG*NH]
    float* cnt   = psq  + (size_t)NG * NH;        // [NG] (pad 512)
    float* stats = cnt  + 512;                    // [128] sum|sumsq
    float* scsh  = stats + 128;                   // [128] scale|shift

    const int TPB = 256;
    const long nElem = (long)NN * NH;             // 3,200,000
    const int elemBlocks = (int)((nElem + TPB - 1) / TPB);
    const long nEdgeElem = (long)NE * NH;         // 51,200,000
    const int edgeBlocks = (int)((nEdgeElem + TPB - 1) / TPB);
    const int gemmBlocks = ((NN / 16) + 1) / 2;   // 3125 row tiles -> 1563 blocks
    const int poolBlocks = (NG * NH + TPB - 1) / TPB;
    const float invM = 1.0f / (float)NN;
    const int embedOff = 0;
    const int stdOff   = 4 * NG * NH;             // second output tensor

    // graph node counts (constant across layers)
    zero_k<<<(NG + TPB - 1) / TPB, TPB, 0, stream>>>(cnt, NG);
    count_k<<<(NN + TPB - 1) / TPB, TPB, 0, stream>>>(batch, cnt, NN);

    // ---- input transform: bufA = x @ Wt + bt ; BN ; pool ----
    gemm_wmma<NF, false><<<gemmBlocks, TPB, 0, stream>>>(x, nullptr, Wt, bt, bufA, NN, 0);
    zero_k<<<1, 128, 0, stream>>>(stats, 128);
    colstats_k<<<250, TPB, 0, stream>>>(bufA, NN, stats);
    bn_prep_k<<<1, 64, 0, stream>>>(stats, gamma0, beta0, scsh, invM);
    zero_k<<<poolBlocks, TPB, 0, stream>>>(psum, (long)NG * NH);
    zero_k<<<poolBlocks, TPB, 0, stream>>>(psq,  (long)NG * NH);
    bn_apply_pool_k<<<elemBlocks, TPB, 0, stream>>>(bufA, batch, scsh, psum, nElem);
    pool_mean_fin_k<<<poolBlocks, TPB, 0, stream>>>(psum, cnt, out + embedOff);
    pool_sq_k<<<elemBlocks, TPB, 0, stream>>>(bufA, batch, psum, psq, nElem);
    pool_std_fin_k<<<poolBlocks, TPB, 0, stream>>>(psq, cnt, out + stdOff);

    // ---- GIN layers ----
    for (int l = 0; l < 3; ++l) {
        const float* w1 = W1 + (size_t)l * NH * NH;
        const float* w2 = W2 + (size_t)l * NH * NH;
        const float* g  = gammas + (size_t)l * NH;
        const float* b  = betas  + (size_t)l * NH;

        zero_k<<<elemBlocks, TPB, 0, stream>>>(bufB, nElem);
        scatter_add_k<<<edgeBlocks, TPB, 0, stream>>>(bufA, src, dst, bufB, nEdgeElem);

        // bufC = relu((h + agg) @ w1)
        gemm_wmma<NH, true><<<gemmBlocks, TPB, 0, stream>>>(bufA, bufB, w1, nullptr, bufC, NN, 1);
        // bufA = relu(bufC @ w2)          (h dead after gemm1, safe to overwrite)
        gemm_wmma<NH, false><<<gemmBlocks, TPB, 0, stream>>>(bufC, nullptr, w2, nullptr, bufA, NN, 1);

        zero_k<<<1, 128, 0, stream>>>(stats, 128);
        colstats_k<<<250, TPB, 0, stream>>>(bufA, NN, stats);
        bn_prep_k<<<1, 64, 0, stream>>>(stats, g, b, scsh, invM);
        zero_k<<<poolBlocks, TPB, 0, stream>>>(psum, (long)NG * NH);
        zero_k<<<poolBlocks, TPB, 0, stream>>>(psq,  (long)NG * NH);
        bn_apply_pool_k<<<elemBlocks, TPB, 0, stream>>>(bufA, batch, scsh, psum, nElem);
        pool_mean_fin_k<<<poolBlocks, TPB, 0, stream>>>(psum, cnt, out + embedOff + (l + 1) * NG * NH);
        pool_sq_k<<<elemBlocks, TPB, 0, stream>>>(bufA, batch, psum, psq, nElem);
        pool_std_fin_k<<<poolBlocks, TPB, 0, stream>>>(psq, cnt, out + stdOff + (l + 1) * NG * NH);
    }
}